// Attention_49632642073127
// MI455X (gfx1250) — compile-verified
//
#include <hip/hip_runtime.h>
#include <hip/hip_bf16.h>
#include <math.h>
#include <stdint.h>

typedef __attribute__((ext_vector_type(2))) float v2f;
typedef __attribute__((ext_vector_type(8))) float v8f;

#define S_LEN 2048
#define HIDDEN 2048
#define NH_Q 16
#define NH_KV 4
#define DH 128

__device__ inline v8f wmma4(v2f a, v2f b, v8f c) {
  // V_WMMA_F32_16X16X4_F32: D = A(16x4 f32) * B(4x16 f32) + C(16x16 f32)
  return __builtin_amdgcn_wmma_f32_16x16x4_f32(
      false, a, false, b, (short)0, c, false, false);
}

__device__ inline v8f vzero8() {
  v8f z = {0.f, 0.f, 0.f, 0.f, 0.f, 0.f, 0.f, 0.f};
  return z;
}

// gfx1250 async global->LDS copy (ASYNCcnt path). 16B per lane.
// dst: LDS byte offset (low 32 bits of a generic shared pointer), src: global.
__device__ inline void async_ld_b128(uint32_t lds_off, const void* gaddr) {
  asm volatile("global_load_async_to_lds_b128 %0, %1, off"
               :
               : "v"(lds_off), "v"(gaddr)
               : "memory");
}
__device__ inline void wait_async0() {
  asm volatile("s_wait_asynccnt 0x0" ::: "memory");
}
__device__ inline uint32_t lds_off32(const void* p) {
  return (uint32_t)(uintptr_t)p;  // LDS aperture: addr[31:0] = LDS offset
}

// ---------------------------------------------------------------------------
// GEMM: out = X[M,K] @ W[K,N], 128x128 tile per workgroup (8 waves), KT=16.
//  - A tile: async global->LDS, row-major, row stride 20 (b64-aligned, banks
//    20*r mod 64 all distinct for r=0..15).
//  - B tile: k-pair interleaved LDS layout [(k>>1)][n][k&1], pair stride 288
//    (288 mod 64 == 32 -> lane-halves hit disjoint bank halves). Fragments
//    are single ds_load_b64, no repack movs.
// mode 0: plain row-major store; mode 1: RoPE + head-major; mode 2: head-major
// ---------------------------------------------------------------------------
#define GTK 16
#define LDA 20
#define LDBP 288  // dwords per k-pair row of B (2*128 + 32)

__global__ __launch_bounds__(256) void gemm_rope_kernel(
    const float* __restrict__ X, const float* __restrict__ W,
    const float* __restrict__ cosT, const float* __restrict__ sinT,
    float* __restrict__ out, int M, int N, int K, int mode, int nheads) {
  __shared__ __align__(16) float As[128 * LDA];
  __shared__ __align__(16) float Bs[(GTK / 2) * LDBP];

  const int tid = threadIdx.x;
  const int lane = tid & 31;
  const int w = tid >> 5;
  const int half = lane >> 4;
  const int l16 = lane & 15;

  const int m0 = blockIdx.y * 128;
  const int n0 = blockIdx.x * 128;

  v8f acc[8];
#pragma unroll
  for (int i = 0; i < 8; ++i) acc[i] = vzero8();

  const int aRow = tid >> 2;        // 0..63 (+64)
  const int aCol = (tid & 3) * 4;   // 0..12
  const int bPair = tid >> 5;       // 0..7 -> k rows 2p, 2p+1
  const int bCol = (tid & 31) * 4;  // 0..124

  const uint32_t aDst0 = lds_off32(&As[aRow * LDA + aCol]);
  const uint32_t aDst1 = lds_off32(&As[(aRow + 64) * LDA + aCol]);

  for (int k0 = 0; k0 < K; k0 += GTK) {
    // A tile: async DMA into LDS
    async_ld_b128(aDst0, &X[(size_t)(m0 + aRow) * K + k0 + aCol]);
    async_ld_b128(aDst1, &X[(size_t)(m0 + aRow + 64) * K + k0 + aCol]);
    // B tile: stage rows 2p, 2p+1 and interleave as k-pairs
    {
      float4 b0 = *(const float4*)&W[(size_t)(k0 + 2 * bPair) * N + n0 + bCol];
      float4 b1 =
          *(const float4*)&W[(size_t)(k0 + 2 * bPair + 1) * N + n0 + bCol];
      const float* e0 = (const float*)&b0;
      const float* e1 = (const float*)&b1;
#pragma unroll
      for (int i = 0; i < 4; ++i) {
        v2f p;
        p.x = e0[i];
        p.y = e1[i];
        *(v2f*)&Bs[bPair * LDBP + (bCol + i) * 2] = p;
      }
    }
    wait_async0();
    __syncthreads();

#pragma unroll
    for (int kk = 0; kk < GTK; kk += 4) {
      v2f afrag = *(const v2f*)&As[(w * 16 + l16) * LDA + kk + half * 2];
#pragma unroll
      for (int nt = 0; nt < 8; ++nt) {
        v2f bfrag =
            *(const v2f*)&Bs[((kk >> 1) + half) * LDBP + (nt * 16 + l16) * 2];
        acc[nt] = wmma4(afrag, bfrag, acc[nt]);
      }
    }
    __syncthreads();
  }

  const int mw = m0 + w * 16;
  if (mode == 0) {
#pragma unroll
    for (int nt = 0; nt < 8; ++nt)
#pragma unroll
      for (int j = 0; j < 8; ++j) {
        int m = mw + j + 8 * half;
        int n = n0 + nt * 16 + l16;
        out[(size_t)m * N + n] = acc[nt][j];
      }
  } else {
    const int h = n0 >> 7;  // 128-wide wave strip == one head
#pragma unroll
    for (int nt = 0; nt < 8; ++nt) {
#pragma unroll
      for (int j = 0; j < 8; ++j) {
        int m = mw + j + 8 * half;
        int b = m >> 11;
        int s = m & (S_LEN - 1);
        int d = nt * 16 + l16;
        float v = acc[nt][j];
        if (mode == 1) {  // q*cos + rotate_half(q)*sin
          float pair = (nt < 4) ? -acc[nt + 4][j] : acc[nt - 4][j];
          float c = cosT[s * DH + d];
          float sn = sinT[s * DH + d];
          v = v * c + pair * sn;
        }
        out[(((size_t)b * nheads + h) * S_LEN + s) * DH + d] = v;
      }
    }
  }
}

// ---------------------------------------------------------------------------
// Flash attention (causal, GQA 4:1). Grid: (S/128, B*H), block 256 (8 waves).
//  - K tile (32x128): async global->LDS, row-major stride 132 (fragments are
//    contiguous k-pairs -> ds_load_b64 directly).
//  - V tile: k-pair interleaved (like GEMM B) -> ds_load_b64 fragments.
//  - P (probs) round-trips through per-wave LDS to reshape C->A layout.
// ---------------------------------------------------------------------------
#define KVT 32
#define LDK 132
#define LDVP 288
#define LDP 36

__global__ __launch_bounds__(256) void flash_attn_kernel(
    const float* __restrict__ Q, const float* __restrict__ Kc,
    const float* __restrict__ Vc, float* __restrict__ O) {
  __shared__ __align__(16) float Ksm[KVT * LDK];
  __shared__ __align__(16) float Vsm[(KVT / 2) * LDVP];
  __shared__ __align__(16) float Psm[8 * 16 * LDP];

  const int tid = threadIdx.x;
  const int lane = tid & 31;
  const int w = tid >> 5;
  const int half = lane >> 4;
  const int l16 = lane & 15;

  const int qb = blockIdx.x * 128;
  const int bh = blockIdx.y;
  const int b = bh >> 4;
  const int h = bh & 15;
  const int hkv = h >> 2;  // N_REP = 4

  const float* Qh = Q + ((size_t)b * NH_Q + h) * S_LEN * DH;
  const float* Kh = Kc + ((size_t)b * NH_KV + hkv) * S_LEN * DH;
  const float* Vh = Vc + ((size_t)b * NH_KV + hkv) * S_LEN * DH;

  // Preload this wave's Q strip (rows qb+w*16+l16, full D) as A-fragments.
  v2f qfrag[32];
  {
    const float* qp = &Qh[(size_t)(qb + w * 16 + l16) * DH + half * 2];
#pragma unroll
    for (int dk = 0; dk < 32; ++dk) {
      qfrag[dk].x = qp[dk * 4];
      qfrag[dk].y = qp[dk * 4 + 1];
    }
  }

  v8f oacc[8];
#pragma unroll
  for (int i = 0; i < 8; ++i) oacc[i] = vzero8();
  float mrow[8], lrow[8];
#pragma unroll
  for (int j = 0; j < 8; ++j) {
    mrow[j] = -1e30f;
    lrow[j] = 0.f;
  }

  float* Pw = &Psm[w * 16 * LDP];
  const float scale = 0.08838834764831843f;  // 1/sqrt(128)

  const int kRow = tid >> 5;        // 0..7 (+8,+16,+24) for K async
  const int kCol = (tid & 31) * 4;  // 0..124
  const int qEnd = qb + 127;

  uint32_t kDst[4];
#pragma unroll
  for (int r = 0; r < 4; ++r)
    kDst[r] = lds_off32(&Ksm[(kRow + r * 8) * LDK + kCol]);

  for (int kv0 = 0; kv0 <= qEnd; kv0 += KVT) {
    // K tile: async DMA into LDS
#pragma unroll
    for (int r = 0; r < 4; ++r) {
      int row = kRow + r * 8;
      async_ld_b128(kDst[r], &Kh[(size_t)(kv0 + row) * DH + kCol]);
    }
    // V tile: stage + k-pair interleave
#pragma unroll
    for (int r = 0; r < 2; ++r) {
      int pair = kRow + r * 8;  // 0..15
      float4 v0 = *(const float4*)&Vh[(size_t)(kv0 + 2 * pair) * DH + kCol];
      float4 v1 = *(const float4*)&Vh[(size_t)(kv0 + 2 * pair + 1) * DH + kCol];
      const float* e0 = (const float*)&v0;
      const float* e1 = (const float*)&v1;
#pragma unroll
      for (int i = 0; i < 4; ++i) {
        v2f p;
        p.x = e0[i];
        p.y = e1[i];
        *(v2f*)&Vsm[pair * LDVP + (kCol + i) * 2] = p;
      }
    }
    wait_async0();
    __syncthreads();

    if (kv0 <= qb + w * 16 + 15) {  // wave-uniform causal skip
      // S = Q @ K^T
      v8f sacc[2];
      sacc[0] = vzero8();
      sacc[1] = vzero8();
#pragma unroll
      for (int dk = 0; dk < 32; ++dk) {
#pragma unroll
        for (int st = 0; st < 2; ++st) {
          v2f kfrag =
              *(const v2f*)&Ksm[(st * 16 + l16) * LDK + dk * 4 + half * 2];
          sacc[st] = wmma4(qfrag[dk], kfrag, sacc[st]);
        }
      }
      // mask + online softmax; row m=j+8*half lives in one 16-lane half
#pragma unroll
      for (int j = 0; j < 8; ++j) {
        int qrow = qb + w * 16 + j + 8 * half;
        float s0 = sacc[0][j] * scale;
        float s1 = sacc[1][j] * scale;
        if (kv0 + l16 > qrow) s0 = -1e30f;
        if (kv0 + 16 + l16 > qrow) s1 = -1e30f;
        float mx = fmaxf(s0, s1);
        mx = fmaxf(mx, __shfl_xor(mx, 1, 32));
        mx = fmaxf(mx, __shfl_xor(mx, 2, 32));
        mx = fmaxf(mx, __shfl_xor(mx, 4, 32));
        mx = fmaxf(mx, __shfl_xor(mx, 8, 32));
        float mnew = fmaxf(mrow[j], mx);
        float corr = __expf(mrow[j] - mnew);
        float p0 = __expf(s0 - mnew);
        float p1 = __expf(s1 - mnew);
        float ts = p0 + p1;
        ts += __shfl_xor(ts, 1, 32);
        ts += __shfl_xor(ts, 2, 32);
        ts += __shfl_xor(ts, 4, 32);
        ts += __shfl_xor(ts, 8, 32);
        lrow[j] = lrow[j] * corr + ts;
        mrow[j] = mnew;
#pragma unroll
        for (int nt = 0; nt < 8; ++nt) oacc[nt][j] *= corr;
        Pw[(j + 8 * half) * LDP + l16] = p0;
        Pw[(j + 8 * half) * LDP + 16 + l16] = p1;
      }
      // O += P @ V (same-wave LDS RAW ordered via DScnt by the compiler)
#pragma unroll
      for (int ks = 0; ks < 8; ++ks) {
        v2f pfrag = *(const v2f*)&Pw[l16 * LDP + ks * 4 + half * 2];
#pragma unroll
        for (int nt = 0; nt < 8; ++nt) {
          v2f vfrag =
              *(const v2f*)&Vsm[(ks * 2 + half) * LDVP + (nt * 16 + l16) * 2];
          oacc[nt] = wmma4(pfrag, vfrag, oacc[nt]);
        }
      }
    }
    __syncthreads();
  }

  // normalize + store O as [B][S][H*D] for the output projection
#pragma unroll
  for (int j = 0; j < 8; ++j) {
    float inv = 1.0f / lrow[j];
    int s = qb + w * 16 + j + 8 * half;
    size_t rowOff = ((size_t)b * S_LEN + s) * HIDDEN + (size_t)h * DH;
#pragma unroll
    for (int nt = 0; nt < 8; ++nt) {
      O[rowOff + nt * 16 + l16] = oacc[nt][j] * inv;
    }
  }
}

// ---------------------------------------------------------------------------
extern "C" void kernel_launch(void* const* d_in, const int* in_sizes, int n_in,
                              void* d_out, int out_size, void* d_ws,
                              size_t ws_size, hipStream_t stream) {
  const float* x = (const float*)d_in[0];
  const float* cosT = (const float*)d_in[1];
  const float* sinT = (const float*)d_in[2];
  const float* Wq = (const float*)d_in[3];
  const float* Wk = (const float*)d_in[4];
  const float* Wv = (const float*)d_in[5];
  const float* Wo = (const float*)d_in[6];
  float* out = (float*)d_out;

  const int M = 2 * S_LEN;  // B*S = 4096
  float* ws = (float*)d_ws;
  float* Qw = ws;                             // [B][16][S][128]
  float* Kw = Qw + (size_t)M * HIDDEN;        // [B][4][S][128]
  float* Vw = Kw + (size_t)M * (NH_KV * DH);  // [B][4][S][128]
  float* Ow = Vw + (size_t)M * (NH_KV * DH);  // [B][S][2048]

  dim3 blk(256);
  gemm_rope_kernel<<<dim3(HIDDEN / 128, M / 128), blk, 0, stream>>>(
      x, Wq, cosT, sinT, Qw, M, HIDDEN, HIDDEN, 1, NH_Q);
  gemm_rope_kernel<<<dim3((NH_KV * DH) / 128, M / 128), blk, 0, stream>>>(
      x, Wk, cosT, sinT, Kw, M, NH_KV * DH, HIDDEN, 1, NH_KV);
  gemm_rope_kernel<<<dim3((NH_KV * DH) / 128, M / 128), blk, 0, stream>>>(
      x, Wv, cosT, sinT, Vw, M, NH_KV * DH, HIDDEN, 2, NH_KV);
  flash_attn_kernel<<<dim3(S_LEN / 128, 2 * NH_Q), blk, 0, stream>>>(Qw, Kw,
                                                                     Vw, Ow);
  gemm_rope_kernel<<<dim3(HIDDEN / 128, M / 128), blk, 0, stream>>>(
      Ow, Wo, cosT, sinT, out, M, HIDDEN, HIDDEN, 0, 0);
}